// DenseLayer_31482110280318
// MI455X (gfx1250) — compile-verified
//
#include <hip/hip_runtime.h>
#include <hip/hip_bf16.h>

typedef __attribute__((ext_vector_type(2))) float v2f;
typedef __attribute__((ext_vector_type(8))) float v8f;
typedef __attribute__((ext_vector_type(4))) int   v4i;

#define NROWS 262144
#define FDIM 512
#define BLOCK 256
#define ROWS_PER_BLOCK 128            // 8 waves * 16 rows
#define KC 32                         // k-chunk (floats)
#define NCHUNK (FDIM / KC)            // 16
#define LDS_STRIDE (KC + 4)           // 36 floats = 144B: 16B-aligned rows, bank-conflict-free reads

#if defined(__has_builtin)
#if __has_builtin(__builtin_amdgcn_global_load_async_to_lds_b128)
#define HAVE_ASYNC_LDS 1
#endif
#endif

#ifdef HAVE_ASYNC_LDS
// Builtin signature (from clang diagnostic): (v4i AS1*, v4i AS3*, imm offset, imm cpol)
typedef __attribute__((address_space(1))) v4i gv4i;   // global int4
typedef __attribute__((address_space(3))) v4i sv4i;   // LDS int4
#endif

__device__ __forceinline__ void wait_async0() {
#if defined(__has_builtin) && __has_builtin(__builtin_amdgcn_s_wait_asynccnt)
  __builtin_amdgcn_s_wait_asynccnt(0);
#else
  asm volatile("s_wait_asynccnt 0x0" ::: "memory");
#endif
}

__global__ __launch_bounds__(BLOCK)
void dense_tanh_wmma(const float* __restrict__ x,
                     const float* __restrict__ w,
                     const float* __restrict__ b,
                     const int*   __restrict__ mask_w,
                     const int*   __restrict__ mask_b,
                     float*       __restrict__ out) {
  __shared__ float wm[FDIM];
  __shared__ __align__(16) float tile[2][ROWS_PER_BLOCK * LDS_STRIDE];

  const int tid      = threadIdx.x;
  const int lane     = tid & 31;
  const int wave     = tid >> 5;
  const int row_base = blockIdx.x * ROWS_PER_BLOCK;

  // Masked weights staged once into LDS (reused by all 16 chunks).
  for (int i = tid; i < FDIM; i += BLOCK) wm[i] = w[i] * (float)mask_w[i];
  const float bm = b[0] * (float)mask_b[0];

  // Cooperative tile fill: 128 rows x 32 floats. 8 consecutive threads cover one
  // row's contiguous 128B (b128 each) -> coalesced streaming of x exactly once.
#ifdef HAVE_ASYNC_LDS
  // CDNA5 path: HBM -> LDS directly, no VGPR staging, tracked by ASYNCcnt.
  auto load_tile = [&](float* dst, int kbase) {
    #pragma unroll
    for (int i = 0; i < (ROWS_PER_BLOCK * KC / 4) / BLOCK; ++i) {   // 4 iters
      const int idx = tid + i * BLOCK;          // 0 .. 1023
      const int row = idx >> 3;                 // 8 b128 segments per row
      const int seg = idx & 7;
      const float* g = x + (size_t)(row_base + row) * FDIM + kbase + seg * 4;
      float*       l = dst + row * LDS_STRIDE + seg * 4;
      __builtin_amdgcn_global_load_async_to_lds_b128(
          (gv4i*)g, (sv4i*)l, /*offset=*/0, /*cpol=*/0);
    }
  };
#else
  auto load_tile = [&](float* dst, int kbase) {
    #pragma unroll
    for (int i = 0; i < (ROWS_PER_BLOCK * KC / 4) / BLOCK; ++i) {   // 4 iters
      const int idx = tid + i * BLOCK;
      const int row = idx >> 3;
      const int seg = idx & 7;
      const float4 v = *(const float4*)(x + (size_t)(row_base + row) * FDIM + kbase + seg * 4);
      *(float4*)(dst + row * LDS_STRIDE + seg * 4) = v;
    }
  };
#endif

  load_tile(tile[0], 0);
#ifdef HAVE_ASYNC_LDS
  wait_async0();
#endif
  __syncthreads();

  // Two accumulators break the WMMA->WMMA RAW chain to distance 2.
  v8f acc0 = {0.f, 0.f, 0.f, 0.f, 0.f, 0.f, 0.f, 0.f};
  v8f acc1 = {0.f, 0.f, 0.f, 0.f, 0.f, 0.f, 0.f, 0.f};
  const int r    = lane & 15;                   // A-fragment row within wave tile
  const int koff = (lane >> 4) * 2;             // lanes 0-15 hold K{0,1}; 16-31 hold K{2,3}

  for (int kc = 0; kc < NCHUNK; ++kc) {
    const int cur = kc & 1;
    if (kc + 1 < NCHUNK) load_tile(tile[cur ^ 1], (kc + 1) * KC);   // overlap with compute

    const float* arow = tile[cur] + (wave * 16 + r) * LDS_STRIDE + koff;
    const float* wrow = wm + kc * KC + koff;
    #pragma unroll
    for (int kk = 0; kk < KC; kk += 8) {
      v2f a0 = {arow[kk],     arow[kk + 1]};     // ds_load_b64, conflict-free
      v2f b0 = {wrow[kk],     wrow[kk + 1]};     // half-wave broadcast of wm
      acc0 = __builtin_amdgcn_wmma_f32_16x16x4_f32(false, a0, false, b0,
                                                   (short)0, acc0, false, false);
      v2f a1 = {arow[kk + 4], arow[kk + 5]};
      v2f b1 = {wrow[kk + 4], wrow[kk + 5]};
      acc1 = __builtin_amdgcn_wmma_f32_16x16x4_f32(false, a1, false, b1,
                                                   (short)0, acc1, false, false);
    }
#ifdef HAVE_ASYNC_LDS
    wait_async0();            // next buffer resident before the barrier
#endif
    __syncthreads();
  }

  const v8f acc = acc0 + acc1;

  // D layout: VGPR j holds row j (lanes 0-15) / row j+8 (lanes 16-31); every
  // column is identical (B was broadcast), so lanes 0 and 16 emit 8 rows each.
  if ((lane & 15) == 0) {
    const int half  = lane >> 4;
    const int rbase = row_base + wave * 16 + half * 8;
    #pragma unroll
    for (int j = 0; j < 8; ++j)
      out[rbase + j] = tanhf(acc[j] + bm);
  }
}

extern "C" void kernel_launch(void* const* d_in, const int* in_sizes, int n_in,
                              void* d_out, int out_size, void* d_ws, size_t ws_size,
                              hipStream_t stream) {
  (void)in_sizes; (void)n_in; (void)out_size; (void)d_ws; (void)ws_size;
  const float* x      = (const float*)d_in[0];
  const float* w      = (const float*)d_in[1];
  const float* b      = (const float*)d_in[2];
  const int*   mask_w = (const int*)d_in[3];
  const int*   mask_b = (const int*)d_in[4];
  float* out = (float*)d_out;

  dim3 grid(NROWS / ROWS_PER_BLOCK);   // 2048 blocks
  dim3 block(BLOCK);
  dense_tanh_wmma<<<grid, block, 0, stream>>>(x, w, b, mask_w, mask_b, out);
}